// DSPA_44976897524187
// MI455X (gfx1250) — compile-verified
//
#include <hip/hip_runtime.h>

// Problem constants
#define N_TOT 4096   // n = D*H*W
#define M_REAL 4094  // conv output length
#define M_PAD 4096   // zero-padded for 16-wide WMMA tiling
#define CCH 64
#define BB 2

typedef float v2f __attribute__((ext_vector_type(2)));
typedef float v8f __attribute__((ext_vector_type(8)));

// D = A(16x4 f32) * B(4x16 f32) + C(16x16 f32)  -- native fp32 WMMA
__device__ __forceinline__ v8f wmma_f32(v2f a, v2f b, v8f c) {
  return __builtin_amdgcn_wmma_f32_16x16x4_f32(false, a, false, b, (short)0, c,
                                               false, false);
}

// ---------------------------------------------------------------------------
// Kernel 1: dilated conv1d (k=3, dilation=2, pad=1) -> x2p[b][c][0..4093], and
// zero for padded columns 4094/4095 so downstream tiles read zeros.
// ---------------------------------------------------------------------------
__global__ __launch_bounds__(256) void conv_kernel(
    const float* __restrict__ x, const float* __restrict__ w,
    const float* __restrict__ bias, float* __restrict__ x2p) {
  int idx = blockIdx.x * blockDim.x + threadIdx.x;  // [0, 2*64*4096)
  int j = idx & (N_TOT - 1);
  int c = (idx >> 12) & (CCH - 1);
  int b = idx >> 18;
  float acc = 0.0f;
  if (j < M_REAL) {
    acc = bias[c];
    const float* xb = x + (size_t)b * CCH * N_TOT;
    const float* wc = w + c * (CCH * 3);
    for (int i = 0; i < CCH; ++i) {
      const float* xi = xb + i * N_TOT;
      float w0 = wc[3 * i + 0];
      float w1 = wc[3 * i + 1];
      float w2 = wc[3 * i + 2];
      float s = w1 * xi[j + 1];          // j+1 <= 4094 < 4096 always valid
      if (j >= 1) s += w0 * xi[j - 1];   // left pad
      if (j + 3 < N_TOT) s += w2 * xi[j + 3];  // right pad (only j=4093 OOB)
      acc += s;
    }
  }
  x2p[idx] = acc;
}

// ---------------------------------------------------------------------------
// Kernel 2: fused  S_b = x1_b^T x2_b  ->  g = 2-way softmax over batch (a
// sigmoid gate, elementwise in (n,m))  ->  O_b += g_b * x2_b^T.
// Each wave: 16 n-rows, both batches. Block: 4 waves = 64 n-rows.
// blockIdx.y splits the m range; partial outputs reduced by kernel 3.
// ---------------------------------------------------------------------------
__global__ __launch_bounds__(128) void attn_kernel(
    const float* __restrict__ x1, const float* __restrict__ x2p,
    float* __restrict__ part, int chunks_per_y) {
  __shared__ float x2s[BB][CCH][17];       // 64x16 tile per batch, pad 17
  __shared__ float gate_s[4][BB][16][18];  // per-wave gate transpose slab

  const int lane = threadIdx.x & 31;
  const int wv = threadIdx.x >> 5;
  const int row = lane & 15;  // M (or N) index within a 16-tile
  const int hi = lane >> 4;   // which K half-pair this lane carries
  const int n0 = blockIdx.x * 64 + wv * 16;

  // A fragments for the score GEMM: A_b[r,c] = x1[b,c,n0+r], K=64 -> 16 subtiles
  v2f a[2][16];
  for (int b = 0; b < 2; ++b)
    for (int t = 0; t < 16; ++t) {
      int c0 = 4 * t + 2 * hi;
      a[b][t].x = x1[((size_t)(b * CCH + c0) * N_TOT) + n0 + row];
      a[b][t].y = x1[((size_t)(b * CCH + c0 + 1) * N_TOT) + n0 + row];
    }

  // Output accumulators: O_b is 16(n) x 64(c) fp32
  v8f o[2][4];
  for (int b = 0; b < 2; ++b)
    for (int u = 0; u < 4; ++u)
      for (int e = 0; e < 8; ++e) o[b][u][e] = 0.0f;

  for (int it = 0; it < chunks_per_y; ++it) {
    int m0 = (blockIdx.y * chunks_per_y + it) * 16;

    __syncthreads();  // previous iteration's readers done before overwrite
    // Cooperative stage of x2[b][0..63][m0..m0+15] (both batches, 8KB)
    for (int k = 0; k < 4; ++k) {
      int q = threadIdx.x + 128 * k;  // 512 float4 units
      int mm4 = q & 3;
      int cc = (q >> 2) & (CCH - 1);
      int bb2 = q >> 8;
      const float4 v =
          *(const float4*)(x2p + ((size_t)(bb2 * CCH + cc) * M_PAD) + m0 + mm4 * 4);
      x2s[bb2][cc][mm4 * 4 + 0] = v.x;
      x2s[bb2][cc][mm4 * 4 + 1] = v.y;
      x2s[bb2][cc][mm4 * 4 + 2] = v.z;
      x2s[bb2][cc][mm4 * 4 + 3] = v.w;
    }
    __syncthreads();

    // ---- scores: S_b (16x16) = A_b (16x64) * x2_b (64x16), 16 chained WMMAs
    v8f s0, s1;
    for (int e = 0; e < 8; ++e) { s0[e] = 0.0f; s1[e] = 0.0f; }
    for (int t = 0; t < 16; ++t) {
      int c0 = 4 * t + 2 * hi;
      v2f b0, b1;
      b0.x = x2s[0][c0][row];     b0.y = x2s[0][c0 + 1][row];
      b1.x = x2s[1][c0][row];     b1.y = x2s[1][c0 + 1][row];
      s0 = wmma_f32(a[0][t], b0, s0);
      s1 = wmma_f32(a[1][t], b1, s1);
    }

    // ---- 2-entry softmax over batch == sigmoid gate (elementwise)
    v8f g0, g1;
    for (int e = 0; e < 8; ++e) {
      float d = s1[e] - s0[e];
      float t0 = 1.0f / (1.0f + __expf(d));  // overflow-safe: inf -> gate 0
      g0[e] = t0;
      g1[e] = 1.0f - t0;
    }
    // C-layout -> A-layout transpose via per-wave LDS (no barrier needed)
    for (int e = 0; e < 8; ++e) {
      gate_s[wv][0][e + 8 * hi][row] = g0[e];
      gate_s[wv][1][e + 8 * hi][row] = g1[e];
    }

    // ---- O_b (16x64) += G_b (16x16) * V_b (16x64), V_b[k,c] = x2[b,c,m0+k]
    for (int t = 0; t < 4; ++t) {
      int k0 = 4 * t + 2 * hi;
      v2f ga0, ga1;
      ga0.x = gate_s[wv][0][row][k0];  ga0.y = gate_s[wv][0][row][k0 + 1];
      ga1.x = gate_s[wv][1][row][k0];  ga1.y = gate_s[wv][1][row][k0 + 1];
      for (int u = 0; u < 4; ++u) {
        v2f vb0, vb1;
        vb0.x = x2s[0][16 * u + row][k0];  vb0.y = x2s[0][16 * u + row][k0 + 1];
        vb1.x = x2s[1][16 * u + row][k0];  vb1.y = x2s[1][16 * u + row][k0 + 1];
        o[0][u] = wmma_f32(ga0, vb0, o[0][u]);
        o[1][u] = wmma_f32(ga1, vb1, o[1][u]);
      }
    }
  }

  // Store partial O for this m-split (C-layout: vgpr e -> n row e+8*hi, lane->c)
  float* pb = part + (size_t)blockIdx.y * (BB * CCH * N_TOT);
  for (int b = 0; b < 2; ++b)
    for (int u = 0; u < 4; ++u)
      for (int e = 0; e < 8; ++e) {
        int c = 16 * u + row;
        int n = n0 + e + 8 * hi;
        pb[(size_t)(b * CCH + c) * N_TOT + n] = o[b][u][e];
      }
}

// ---------------------------------------------------------------------------
// Kernel 3: out = x1 + sum over m-splits (deterministic reduction, no atomics)
// ---------------------------------------------------------------------------
__global__ __launch_bounds__(256) void reduce_kernel(
    const float* __restrict__ x1, const float* __restrict__ part,
    float* __restrict__ out, int msplit) {
  int idx = blockIdx.x * blockDim.x + threadIdx.x;
  float acc = x1[idx];
  for (int y = 0; y < msplit; ++y)
    acc += part[(size_t)y * (BB * CCH * N_TOT) + idx];
  out[idx] = acc;
}

extern "C" void kernel_launch(void* const* d_in, const int* in_sizes, int n_in,
                              void* d_out, int out_size, void* d_ws,
                              size_t ws_size, hipStream_t stream) {
  (void)in_sizes; (void)n_in; (void)out_size;
  const float* x = (const float*)d_in[0];       // [2,64,16,16,16] fp32
  const float* w = (const float*)d_in[1];       // [64,64,3] fp32
  const float* bias = (const float*)d_in[2];    // [64] fp32
  float* out = (float*)d_out;                   // [2,64,4096] fp32

  const size_t plane = (size_t)BB * CCH * N_TOT;  // 524288 elems (2 MB)
  float* x2p = (float*)d_ws;
  float* part = x2p + plane;

  // m-split sized to workspace (deterministic given ws_size)
  size_t avail = ws_size / sizeof(float);
  int msplit = 8;
  while (msplit > 1 && plane + (size_t)msplit * plane > avail) msplit >>= 1;
  int chunks_per_y = (M_PAD / 16) / msplit;  // 256 / msplit

  conv_kernel<<<(int)(plane / 256), 256, 0, stream>>>(x, w, bias, x2p);

  dim3 grid(N_TOT / 64, msplit);
  attn_kernel<<<grid, 128, 0, stream>>>(x, x2p, part, chunks_per_y);

  reduce_kernel<<<(int)(plane / 256), 256, 0, stream>>>(x, part, out, msplit);
}